// GCNConv_58411555225969
// MI455X (gfx1250) — compile-verified
//
#include <hip/hip_runtime.h>
#include <hip/hip_bf16.h>

typedef __attribute__((ext_vector_type(2))) float v2f;
typedef __attribute__((ext_vector_type(8))) float v8f;

#define CH 128   // IN_CH == OUT_CH == 128
#define XP 132   // padded LDS row stride (dwords): 132 % 64 = 4 -> conflict-free,
                 // 132*4 B keeps float4 (16 B) staging alignment

// ---------------------------------------------------------------------------
// 1) deg = 0
__global__ __launch_bounds__(256) void deg_zero_kernel(float* __restrict__ deg, int n) {
    int i = blockIdx.x * blockDim.x + threadIdx.x;
    if (i < n) deg[i] = 0.0f;
}

// 2) deg[row[e]] += 1  (scatter count)
__global__ __launch_bounds__(256) void deg_count_kernel(const int* __restrict__ row,
                                                        float* __restrict__ deg, int n_edges) {
    int stride = gridDim.x * blockDim.x;
    for (int e = blockIdx.x * blockDim.x + threadIdx.x; e < n_edges; e += stride) {
        atomicAdd(&deg[row[e]], 1.0f);
    }
}

// 3) deg <- deg^-1/2 (in place). Every node has a self loop so deg >= 1.
__global__ __launch_bounds__(256) void deg_rsqrt_kernel(float* __restrict__ deg, int n) {
    int i = blockIdx.x * blockDim.x + threadIdx.x;
    if (i < n) deg[i] = rsqrtf(deg[i]);
}

// 4) out[n][c] = bias[c]  (also un-poisons d_out)
__global__ __launch_bounds__(256) void out_init_kernel(const float* __restrict__ bias,
                                                       float* __restrict__ out, long long total) {
    long long i = (long long)blockIdx.x * blockDim.x + threadIdx.x;
    if (i < total) out[i] = bias[i & (CH - 1)];
}

// 5) wt = W^T (so each lane's B pair {W[ka][n], W[ka+1][n]} becomes contiguous)
__global__ __launch_bounds__(256) void transpose_w_kernel(const float* __restrict__ w,
                                                          float* __restrict__ wt) {
    int i = blockIdx.x * blockDim.x + threadIdx.x;  // 0 .. 128*128-1
    int r = i >> 7, c = i & (CH - 1);
    wt[c * CH + r] = w[i];
}

// ---------------------------------------------------------------------------
// 6) support = x @ W via V_WMMA_F32_16X16X4_F32 (fp32 tensor path, wave32).
//    One block = 16 rows x 128 cols; 8 waves, one 16x16 col tile per wave.
//    x-tile (16 rows) and all of W^T staged in padded LDS (~74 KB; 4 blocks
//    per 320 KB WGP). Hot loop is pure ds_load_b64 + v_wmma.
//    Two accumulators break the WMMA->WMMA dependency chain in half.
//    n_nodes (100000) is a multiple of 16 -> exact tiles, EXEC all-ones.
__global__ __launch_bounds__(256) void gcn_gemm_wmma_kernel(const float* __restrict__ x,
                                                            const float* __restrict__ wt,
                                                            float* __restrict__ support) {
    __shared__ float sX[16 * XP];       // 8.25 KB (padded x-tile)
    __shared__ float sWt[CH * XP];      // 66 KB  (padded W^T)

    const int row0 = blockIdx.x * 16;
    const int tid  = threadIdx.x;

    // Stage x-tile: 512 float4 / 256 threads = 2 each (row-major -> padded rows)
    {
        const float4* xv = (const float4*)(x + (long long)row0 * CH);
        #pragma unroll
        for (int i = 0; i < 2; ++i) {
            int f = tid + i * 256;              // float4 id
            int r = f >> 5, c4 = f & 31;        // 32 float4 per row
            *(float4*)&sX[r * XP + c4 * 4] = xv[f];
        }
    }
    // Stage W^T: 4096 float4 / 256 threads = 16 each
    {
        const float4* wv = (const float4*)wt;
        #pragma unroll
        for (int i = 0; i < 16; ++i) {
            int f = tid + i * 256;
            int r = f >> 5, c4 = f & 31;
            *(float4*)&sWt[r * XP + c4 * 4] = wv[f];
        }
    }
    __syncthreads();

    const int wave  = tid >> 5;        // 0..7 -> 16-col tile
    const int lane  = tid & 31;
    const int col0  = wave * 16;
    const int m     = lane & 15;       // M (for A) / N (for B,C,D)
    const int khalf = lane >> 4;       // 0: K={k0,k0+1}, 1: K={k0+2,k0+3}

    const float* aRow = &sX[m * XP + 2 * khalf];
    const float* bRow = &sWt[(col0 + m) * XP + 2 * khalf];

    v8f acc0 = {}, acc1 = {};
    #pragma unroll
    for (int k0 = 0; k0 < CH / 2; k0 += 4) {
        // first K half -> acc0
        v2f a0 = *(const v2f*)(aRow + k0);
        v2f b0 = *(const v2f*)(bRow + k0);
        acc0 = __builtin_amdgcn_wmma_f32_16x16x4_f32(false, a0, false, b0,
                                                     (short)0, acc0, false, false);
        // second K half -> acc1 (independent chain)
        v2f a1 = *(const v2f*)(aRow + k0 + CH / 2);
        v2f b1 = *(const v2f*)(bRow + k0 + CH / 2);
        acc1 = __builtin_amdgcn_wmma_f32_16x16x4_f32(false, a1, false, b1,
                                                     (short)0, acc1, false, false);
    }

    // C/D layout: VGPR i -> M = i + 8*khalf, N = m
    float* outp = support + (long long)(row0 + 8 * khalf) * CH + col0 + m;
    #pragma unroll
    for (int i = 0; i < 8; ++i) {
        outp[(long long)i * CH] = acc0[i] + acc1[i];
    }
}

// ---------------------------------------------------------------------------
// 7) Edge scatter: one wave32 per edge; each lane owns 4 consecutive channels.
//    out[row] += (dis[row]*dis[col]) * support[col]; fp32 atomics resolve in
//    L2 (support + out both fit in the 192 MB L2). Next edge's support row is
//    prefetched (global_prefetch_b8) to hide gather latency behind atomics.
__global__ __launch_bounds__(256) void gcn_scatter_kernel(const int* __restrict__ row,
                                                          const int* __restrict__ col,
                                                          const float* __restrict__ dis,
                                                          const float* __restrict__ support,
                                                          float* __restrict__ out,
                                                          int n_edges) {
    const int lane   = threadIdx.x & 31;
    const int waveId = (blockIdx.x * blockDim.x + threadIdx.x) >> 5;
    const int nWaves = (gridDim.x * blockDim.x) >> 5;

    for (int e = waveId; e < n_edges; e += nWaves) {
        const int r = row[e];
        const int c = col[e];
        // prefetch the next edge's gather row while this edge's atomics issue
        if (e + nWaves < n_edges) {
            const int cn = col[e + nWaves];
            __builtin_prefetch(support + (long long)cn * CH + lane * 4, 0, 0);
        }
        const float nrm = dis[r] * dis[c];
        const float4 v = ((const float4*)(support + (long long)c * CH))[lane];
        float* dst = out + (long long)r * CH + lane * 4;
        atomicAdd(dst + 0, nrm * v.x);
        atomicAdd(dst + 1, nrm * v.y);
        atomicAdd(dst + 2, nrm * v.z);
        atomicAdd(dst + 3, nrm * v.w);
    }
}

// ---------------------------------------------------------------------------
extern "C" void kernel_launch(void* const* d_in, const int* in_sizes, int n_in,
                              void* d_out, int out_size, void* d_ws, size_t ws_size,
                              hipStream_t stream) {
    const float* x    = (const float*)d_in[0];
    const int*   edge = (const int*)d_in[1];   // [2, E] flat
    const float* w    = (const float*)d_in[2]; // [128,128]
    const float* bias = (const float*)d_in[3]; // [128]

    const int n_nodes = in_sizes[0] / CH;      // 100000
    const int n_edges = in_sizes[1] / 2;       // 1700000
    const int* rowp = edge;
    const int* colp = edge + n_edges;

    float* support = (float*)d_ws;                              // n_nodes*128 floats
    float* deg     = support + (long long)n_nodes * CH;         // n_nodes floats
    float* wt      = deg + n_nodes;                             // 128*128 floats
    float* out     = (float*)d_out;

    // 1) deg = 0
    deg_zero_kernel<<<(n_nodes + 255) / 256, 256, 0, stream>>>(deg, n_nodes);
    // 2) degree count
    deg_count_kernel<<<2048, 256, 0, stream>>>(rowp, deg, n_edges);
    // 3) deg^-1/2
    deg_rsqrt_kernel<<<(n_nodes + 255) / 256, 256, 0, stream>>>(deg, n_nodes);
    // 4) out = bias (broadcast)
    {
        long long total = (long long)n_nodes * CH;
        int blocks = (int)((total + 255) / 256);
        out_init_kernel<<<blocks, 256, 0, stream>>>(bias, out, total);
    }
    // 5) wt = W^T
    transpose_w_kernel<<<(CH * CH) / 256, 256, 0, stream>>>(w, wt);
    // 6) support = x @ W  (WMMA fp32, LDS-resident operands)
    gcn_gemm_wmma_kernel<<<n_nodes / 16, 256, 0, stream>>>(x, wt, support);
    // 7) scatter-add messages
    gcn_scatter_kernel<<<4096, 256, 0, stream>>>(rowp, colp, deg, support, out, n_edges);
}